// DARPDecoder_77807627534729
// MI455X (gfx1250) — compile-verified
//
#include <hip/hip_runtime.h>
#include <hip/hip_bf16.h>
#include <math.h>

typedef __attribute__((ext_vector_type(2))) float v2f;
typedef __attribute__((ext_vector_type(8))) float v8f;

#define B_DIM 1024
#define N_DIM 512
#define D_DIM 128
#define K_CAT 520   // 4*128 + 3 state feats, padded to multiple of 4 (K-step)

// ---------------------------------------------------------------------------
// Kernel 1: per-batch reductions + gathers -> packed X row of length 520:
// [h_current | h_first | graph_emb | h_visited | cap, time, step, 0-pad]
// ---------------------------------------------------------------------------
__global__ __launch_bounds__(256) void prep_kernel(
    const float* __restrict__ node_emb,
    const float* __restrict__ current_time,
    const float* __restrict__ used_cap,
    const float* __restrict__ veh_cap,
    const int*   __restrict__ current_node,
    const int*   __restrict__ previous_action,
    const int*   __restrict__ first_node,
    const unsigned char* __restrict__ visited,
    const int*   __restrict__ step_i,
    float* __restrict__ X)
{
    const int b   = blockIdx.x;
    const int tid = threadIdx.x;
    const int d4  = tid & 31;   // which float4 chunk of D
    const int g   = tid >> 5;   // row-group 0..7

    __shared__ float lds_all[8][D_DIM];
    __shared__ float lds_vis[8][D_DIM];
    __shared__ float lds_cnt[8];

    const float* base = node_emb + (size_t)b * N_DIM * D_DIM;
    float4 sa = {0.f, 0.f, 0.f, 0.f};
    float4 sv = {0.f, 0.f, 0.f, 0.f};
    float  cnt = 0.f;

    for (int n = g; n < N_DIM; n += 8) {
        float4 v = *(const float4*)(base + n * D_DIM + d4 * 4);
        sa.x += v.x; sa.y += v.y; sa.z += v.z; sa.w += v.w;
        bool vis = visited[b * N_DIM + n] != 0;
        if (vis) {
            sv.x += v.x; sv.y += v.y; sv.z += v.z; sv.w += v.w;
            if (d4 == 0) cnt += 1.f;
        }
    }
    ((float4*)&lds_all[g][0])[d4] = sa;
    ((float4*)&lds_vis[g][0])[d4] = sv;
    if (d4 == 0) lds_cnt[g] = cnt;
    __syncthreads();

    if (tid < D_DIM) {
        float s = 0.f, s2 = 0.f, c = 0.f;
        #pragma unroll
        for (int gg = 0; gg < 8; ++gg) {
            s  += lds_all[gg][tid];
            s2 += lds_vis[gg][tid];
            c  += lds_cnt[gg];
        }
        c = fmaxf(c, 1.f);

        int cur  = current_node[b];
        int prev = previous_action[b];
        int fn   = first_node[b];
        if (prev == 0 && cur != 0) fn = cur;   // left_depot
        if (cur == 0)              fn = 0;

        float* xr = X + (size_t)b * K_CAT;
        xr[tid]              = base[cur * D_DIM + tid];   // h_current
        xr[D_DIM + tid]      = base[fn  * D_DIM + tid];   // h_first
        xr[2 * D_DIM + tid]  = s * (1.f / (float)N_DIM);  // graph_emb
        xr[3 * D_DIM + tid]  = s2 / c;                    // h_visited
        if (tid == 0) {
            xr[512] = veh_cap[b] - used_cap[b];
            xr[513] = current_time[b] * (1.f / 1440.f);
            xr[514] = (float)step_i[b] * (1.f / (float)(2 * N_DIM));
            xr[515] = 0.f; xr[516] = 0.f; xr[517] = 0.f; xr[518] = 0.f; xr[519] = 0.f;
        }
    }
}

// ---------------------------------------------------------------------------
// Kernel 2: stack the 5 weight matrices into Wcat (520 x 128), zero padding.
// ---------------------------------------------------------------------------
__global__ __launch_bounds__(256) void wcat_kernel(
    const float* __restrict__ Wl, const float* __restrict__ Wf,
    const float* __restrict__ Wg, const float* __restrict__ Wv,
    const float* __restrict__ Ws, float* __restrict__ Wcat)
{
    int idx = blockIdx.x * 256 + threadIdx.x;           // 520*128 = 66560
    if (idx >= K_CAT * D_DIM) return;
    int r = idx >> 7, c = idx & 127;
    float v;
    if      (r < 128) v = Wl[r * 128 + c];
    else if (r < 256) v = Wf[(r - 128) * 128 + c];
    else if (r < 384) v = Wg[(r - 256) * 128 + c];
    else if (r < 512) v = Wv[(r - 384) * 128 + c];
    else if (r < 515) v = Ws[(r - 512) * 128 + c];
    else              v = 0.f;
    Wcat[idx] = v;
}

// ---------------------------------------------------------------------------
// Kernel 3: q = X (1024x520) @ Wcat (520x128) + b_state, full-fp32 WMMA.
// One wave32 per 16x16 C tile; V_WMMA_F32_16X16X4_F32, 130 K-steps.
// Layout (ISA 7.12.2): A/B element k = 2*(lane>>4)+vgpr, 16-dim = lane&15;
// C/D row m = vgpr + 8*(lane>>4), col = lane&15.
// ---------------------------------------------------------------------------
__global__ __launch_bounds__(128) void gemm_q_kernel(
    const float* __restrict__ X, const float* __restrict__ Wcat,
    const float* __restrict__ b_state, float* __restrict__ q)
{
    const int wave  = (blockIdx.x * 128 + threadIdx.x) >> 5;  // 0..511
    const int lane  = threadIdx.x & 31;
    const int mtile = wave >> 3;                              // 0..63
    const int ntile = wave & 7;                               // 0..7
    const int row   = mtile * 16 + (lane & 15);
    const int col   = ntile * 16 + (lane & 15);
    const int khalf = (lane >> 4) << 1;                       // 0 or 2

    v8f c;
    const float bias = b_state[col];
    #pragma unroll
    for (int v = 0; v < 8; ++v) c[v] = bias;

    const float* xrow = X + (size_t)row * K_CAT + khalf;
    for (int k0 = 0; k0 < K_CAT; k0 += 4) {
        v2f a, bm;
        a[0]  = xrow[k0 + 0];
        a[1]  = xrow[k0 + 1];
        bm[0] = Wcat[(k0 + khalf + 0) * D_DIM + col];
        bm[1] = Wcat[(k0 + khalf + 1) * D_DIM + col];
        c = __builtin_amdgcn_wmma_f32_16x16x4_f32(
                false, a, false, bm, (short)0, c, false, false);
    }
    const int rbase = mtile * 16 + ((lane >> 4) << 3);
    #pragma unroll
    for (int v = 0; v < 8; ++v)
        q[(size_t)(rbase + v) * D_DIM + col] = c[v];
}

// ---------------------------------------------------------------------------
// Kernel 4: qk = q (1024x128) @ W_key^T (128x128). B[k][n] = W_key[n][k]
// so both operands are contiguous float2 loads per lane.
// ---------------------------------------------------------------------------
__global__ __launch_bounds__(128) void gemm_qk_kernel(
    const float* __restrict__ q, const float* __restrict__ Wkey,
    float* __restrict__ qk)
{
    const int wave  = (blockIdx.x * 128 + threadIdx.x) >> 5;
    const int lane  = threadIdx.x & 31;
    const int mtile = wave >> 3;
    const int ntile = wave & 7;
    const int row   = mtile * 16 + (lane & 15);
    const int col   = ntile * 16 + (lane & 15);
    const int khalf = (lane >> 4) << 1;

    v8f c;
    #pragma unroll
    for (int v = 0; v < 8; ++v) c[v] = 0.f;

    const float* qrow = q    + (size_t)row * D_DIM + khalf;
    const float* wrow = Wkey + (size_t)col * D_DIM + khalf;   // transposed B
    for (int k0 = 0; k0 < D_DIM; k0 += 4) {
        v2f a, bm;
        a[0]  = qrow[k0 + 0];
        a[1]  = qrow[k0 + 1];
        bm[0] = wrow[k0 + 0];
        bm[1] = wrow[k0 + 1];
        c = __builtin_amdgcn_wmma_f32_16x16x4_f32(
                false, a, false, bm, (short)0, c, false, false);
    }
    const int rbase = mtile * 16 + ((lane >> 4) << 3);
    #pragma unroll
    for (int v = 0; v < 8; ++v)
        qk[(size_t)(rbase + v) * D_DIM + col] = c[v];
}

// ---------------------------------------------------------------------------
// Kernel 5: score[b,n] = node_emb[b,n]·qk[b]/sqrt(D) - travel penalty,
// tanh clip, mask, block-wide log_softmax over N=512. One block per b.
// ---------------------------------------------------------------------------
__global__ __launch_bounds__(512) void score_kernel(
    const float* __restrict__ node_emb, const float* __restrict__ qk,
    const float* __restrict__ ttm, const int* __restrict__ current_node,
    const unsigned char* __restrict__ action_mask,
    const int* __restrict__ h3, float* __restrict__ out)
{
    const int b    = blockIdx.x;
    const int tid  = threadIdx.x;
    const int lane = tid & 31;
    const int w    = tid >> 5;                // 16 waves

    __shared__ float sq[D_DIM];
    __shared__ float sc[N_DIM];
    __shared__ float red[N_DIM];

    if (tid < D_DIM) sq[tid] = qk[b * D_DIM + tid];
    __syncthreads();

    const float* base = node_emb + (size_t)b * N_DIM * D_DIM;
    const int cur   = current_node[b];
    const int rbase = h3[b * N_DIM + cur] * N_DIM;

    const float q0 = sq[lane * 4 + 0], q1 = sq[lane * 4 + 1];
    const float q2 = sq[lane * 4 + 2], q3 = sq[lane * 4 + 3];

    const float inv_sqrt_d  = 0.08838834764831845f;              // 1/sqrt(128)
    const float travel_scl  = 1.f / (1440.f * 1.41421356237f);

    for (int n = w; n < N_DIM; n += 16) {
        float4 v = *(const float4*)(base + n * D_DIM + lane * 4);
        float p = v.x * q0 + v.y * q1 + v.z * q2 + v.w * q3;
        #pragma unroll
        for (int m = 16; m > 0; m >>= 1) p += __shfl_xor(p, m, 32);
        if (lane == 0) {
            float s = p * inv_sqrt_d;
            s -= ttm[rbase + h3[b * N_DIM + n]] * travel_scl;
            s  = 10.f * tanhf(s * 0.1f);
            if (!action_mask[b * N_DIM + n]) s = -100000000.0f;
            sc[n] = s;
        }
    }
    __syncthreads();

    // max reduction
    red[tid] = sc[tid];
    __syncthreads();
    for (int s = 256; s > 0; s >>= 1) {
        if (tid < s) red[tid] = fmaxf(red[tid], red[tid + s]);
        __syncthreads();
    }
    const float maxv = red[0];
    __syncthreads();

    // sum of exp
    red[tid] = expf(sc[tid] - maxv);
    __syncthreads();
    for (int s = 256; s > 0; s >>= 1) {
        if (tid < s) red[tid] += red[tid + s];
        __syncthreads();
    }
    const float logZ = logf(red[0]);
    out[(size_t)b * N_DIM + tid] = sc[tid] - maxv - logZ;
}

// ---------------------------------------------------------------------------
extern "C" void kernel_launch(void* const* d_in, const int* in_sizes, int n_in,
                              void* d_out, int out_size, void* d_ws, size_t ws_size,
                              hipStream_t stream) {
    const float* node_emb   = (const float*)d_in[0];
    const float* W_last     = (const float*)d_in[1];
    const float* W_first    = (const float*)d_in[2];
    const float* W_graph    = (const float*)d_in[3];
    const float* W_visited  = (const float*)d_in[4];
    const float* W_key      = (const float*)d_in[5];
    const float* W_state    = (const float*)d_in[6];
    const float* b_state    = (const float*)d_in[7];
    const float* cur_time   = (const float*)d_in[8];
    const float* used_cap   = (const float*)d_in[9];
    const float* veh_cap    = (const float*)d_in[10];
    const float* ttm        = (const float*)d_in[11];
    const int*   cur_node   = (const int*)d_in[12];
    const int*   prev_act   = (const int*)d_in[13];
    const int*   first_node = (const int*)d_in[14];
    const unsigned char* visited = (const unsigned char*)d_in[15];
    const unsigned char* amask   = (const unsigned char*)d_in[16];
    const int*   step_i     = (const int*)d_in[17];
    const int*   h3         = (const int*)d_in[18];
    float* out = (float*)d_out;

    // workspace layout (floats)
    float* X    = (float*)d_ws;                 // 1024*520
    float* Wcat = X    + (size_t)B_DIM * K_CAT; // 520*128
    float* q    = Wcat + (size_t)K_CAT * D_DIM; // 1024*128
    float* qk   = q    + (size_t)B_DIM * D_DIM; // 1024*128

    prep_kernel<<<B_DIM, 256, 0, stream>>>(node_emb, cur_time, used_cap, veh_cap,
                                           cur_node, prev_act, first_node,
                                           visited, step_i, X);
    wcat_kernel<<<(K_CAT * D_DIM + 255) / 256, 256, 0, stream>>>(
        W_last, W_first, W_graph, W_visited, W_state, Wcat);
    gemm_q_kernel<<<128, 128, 0, stream>>>(X, Wcat, b_state, q);
    gemm_qk_kernel<<<128, 128, 0, stream>>>(q, W_key, qk);
    score_kernel<<<B_DIM, 512, 0, stream>>>(node_emb, qk, ttm, cur_node,
                                            amask, h3, out);
}